// CALayer_31447750541614
// MI455X (gfx1250) — compile-verified
//
#include <hip/hip_runtime.h>
#include <hip/hip_bf16.h>
#include <math.h>

typedef __attribute__((ext_vector_type(2))) float v2f;
typedef __attribute__((ext_vector_type(8))) float v8f;

#define CH    64
#define RED   16
#define BATCH 16
#define HW    65536           // 256*256
#define HW4   16384           // HW / 4
#define NPLANE (BATCH * CH)   // 1024

// ---------------------------------------------------------------------------
// Kernel 1: per-(b,c) plane sum + max reduction. One block per plane.
// Planes are contiguous (x is [B,C,H,W] row-major) -> perfectly coalesced
// float4 streaming reads. Result: ws[0:1024)=sum, ws[1024:2048)=max.
// ---------------------------------------------------------------------------
__global__ __launch_bounds__(256) void ca_pool_kernel(const float* __restrict__ x,
                                                      float* __restrict__ ws) {
    const int plane = blockIdx.x;                      // b*64 + c
    const float4* p4 = (const float4*)(x + (size_t)plane * HW);

    float s = 0.0f;
    float m = -INFINITY;
    for (int i = threadIdx.x; i < HW4; i += 256) {
        float4 v = p4[i];
        s += (v.x + v.y) + (v.z + v.w);
        m = fmaxf(m, fmaxf(fmaxf(v.x, v.y), fmaxf(v.z, v.w)));
    }

    // wave32 reduction
    #pragma unroll
    for (int off = 16; off > 0; off >>= 1) {
        s += __shfl_down(s, off, 32);
        m = fmaxf(m, __shfl_down(m, off, 32));
    }

    __shared__ float ssum[8];
    __shared__ float smax[8];
    const int wid = threadIdx.x >> 5;
    if ((threadIdx.x & 31) == 0) { ssum[wid] = s; smax[wid] = m; }
    __syncthreads();

    if (threadIdx.x == 0) {
        float S = ssum[0], M = smax[0];
        #pragma unroll
        for (int w = 1; w < 8; ++w) { S += ssum[w]; M = fmaxf(M, smax[w]); }
        ws[plane]          = S;   // sum (scaled to mean in kernel 2)
        ws[NPLANE + plane] = M;   // max
    }
}

// ---------------------------------------------------------------------------
// Kernel 2: shared 2-layer MLP on v_avg and v_max via V_WMMA_F32_16X16X4_F32.
// One wave (32 lanes), EXEC all ones.
//   H = relu(V[16x64] x W1^T[64x16] + b1)   : M=16,N=16,K=64 -> 16 wmma steps
//   Z += H[16x16]  x W2^T[16x64] + b2       : M=16,N=64,K=16 -> 4 tiles x 4 steps
// done for both pooled vectors, then y = sigmoid(Z) -> ws[2048:3072).
// VGPR layouts per CDNA5 ISA 7.12.2:
//   A 16x4 f32 : lanes 0-15 -> {K=k0,k0+1}, lanes 16-31 -> {K=k0+2,k0+3}, M=lane&15
//   B 4x16 f32 : same K split, N = lane&15
//   C/D 16x16  : vgpr i -> M = i + 8*(lane>=16), N = lane&15
// ---------------------------------------------------------------------------
__global__ __launch_bounds__(32) void ca_mlp_kernel(const float* __restrict__ w1,
                                                    const float* __restrict__ b1,
                                                    const float* __restrict__ w2,
                                                    const float* __restrict__ b2,
                                                    float* __restrict__ ws) {
    __shared__ float V[BATCH * CH];     // current pooled vectors
    __shared__ float H[BATCH * RED];    // hidden activations
    __shared__ float Z[BATCH * CH];     // accumulated pre-sigmoid logits

    const int lane = threadIdx.x;       // 0..31
    const int nlo  = lane & 15;
    const int hi   = (lane >> 4) & 1;   // 0 for lanes 0-15, 1 for 16-31

    for (int i = lane; i < BATCH * CH; i += 32) Z[i] = 0.0f;

    for (int pass = 0; pass < 2; ++pass) {
        const float* src   = ws + pass * NPLANE;
        const float  scale = (pass == 0) ? (1.0f / (float)HW) : 1.0f;

        __syncthreads();
        for (int i = lane; i < BATCH * CH; i += 32) V[i] = src[i] * scale;
        __syncthreads();

        // ---- Stage 1: H = relu(V x W1^T + b1), K = 64 ----
        v8f acc;
        #pragma unroll
        for (int i = 0; i < 8; ++i) acc[i] = b1[nlo];

        #pragma unroll
        for (int ki = 0; ki < 16; ++ki) {
            const int k0 = ki * 4 + hi * 2;
            v2f a, b;
            a.x = V[nlo * CH + k0];         // A[m=nlo][k0]
            a.y = V[nlo * CH + k0 + 1];
            b.x = w1[nlo * CH + k0];        // B[k0][n=nlo] = w1[n][k0]
            b.y = w1[nlo * CH + k0 + 1];
            acc = __builtin_amdgcn_wmma_f32_16x16x4_f32(
                false, a, false, b, (short)0, acc, false, false);
        }

        #pragma unroll
        for (int i = 0; i < 8; ++i) {
            const int m = i + hi * 8;
            H[m * RED + nlo] = fmaxf(acc[i], 0.0f);
        }
        __syncthreads();

        // ---- Stage 2: Z += H x W2^T + b2, K = 16, four N-tiles of 16 ----
        #pragma unroll
        for (int t = 0; t < 4; ++t) {
            const int n0 = t * 16;
            v8f zacc;
            #pragma unroll
            for (int i = 0; i < 8; ++i) zacc[i] = b2[n0 + nlo];

            #pragma unroll
            for (int ki = 0; ki < 4; ++ki) {
                const int k0 = ki * 4 + hi * 2;
                v2f a, b;
                a.x = H[nlo * RED + k0];            // A[m=nlo][k=r]
                a.y = H[nlo * RED + k0 + 1];
                b.x = w2[(n0 + nlo) * RED + k0];    // B[k][n] = w2[n][k]
                b.y = w2[(n0 + nlo) * RED + k0 + 1];
                zacc = __builtin_amdgcn_wmma_f32_16x16x4_f32(
                    false, a, false, b, (short)0, zacc, false, false);
            }

            #pragma unroll
            for (int i = 0; i < 8; ++i) {
                const int m = i + hi * 8;
                Z[m * CH + n0 + nlo] += zacc[i];
            }
        }
        __syncthreads();
    }

    // y = sigmoid(Z)
    float* y = ws + 2 * NPLANE;
    for (int i = lane; i < BATCH * CH; i += 32) {
        y[i] = 1.0f / (1.0f + __expf(-Z[i]));
    }
}

// ---------------------------------------------------------------------------
// Kernel 3: out = x * y[plane], grid-stride float4 streaming. Bandwidth bound.
// ---------------------------------------------------------------------------
__global__ __launch_bounds__(256) void ca_scale_kernel(const float* __restrict__ x,
                                                       const float* __restrict__ y,
                                                       float* __restrict__ out,
                                                       long long n4) {
    const float4* x4 = (const float4*)x;
    float4* o4 = (float4*)out;
    long long i = (long long)blockIdx.x * blockDim.x + threadIdx.x;
    const long long stride = (long long)gridDim.x * blockDim.x;
    for (; i < n4; i += stride) {
        const int plane = (int)(i >> 14);   // (i*4) / 65536 ; float4 never crosses a plane
        const float sc = y[plane];
        float4 v = x4[i];
        v.x *= sc; v.y *= sc; v.z *= sc; v.w *= sc;
        o4[i] = v;
    }
}

extern "C" void kernel_launch(void* const* d_in, const int* in_sizes, int n_in,
                              void* d_out, int out_size, void* d_ws, size_t ws_size,
                              hipStream_t stream) {
    const float* x  = (const float*)d_in[0];   // [16,64,256,256]
    const float* w1 = (const float*)d_in[1];   // [16,64]
    const float* b1 = (const float*)d_in[2];   // [16]
    const float* w2 = (const float*)d_in[3];   // [64,16]
    const float* b2 = (const float*)d_in[4];   // [64]
    float* out = (float*)d_out;
    float* ws  = (float*)d_ws;                 // 3*1024 floats used

    ca_pool_kernel<<<NPLANE, 256, 0, stream>>>(x, ws);
    ca_mlp_kernel<<<1, 32, 0, stream>>>(w1, b1, w2, b2, ws);

    const long long n4 = (long long)NPLANE * HW4;  // 16,777,216 float4s
    ca_scale_kernel<<<16384, 256, 0, stream>>>(x, ws + 2 * NPLANE, out, n4);
}